// MaskedCRFLoss_41936060678703
// MI455X (gfx1250) — compile-verified
//
#include <hip/hip_runtime.h>
#include <math.h>

typedef float v2f __attribute__((ext_vector_type(2)));
typedef float v8f __attribute__((ext_vector_type(8)));

#define S_LEN 512
#define B_DIM 1024
#define T_DIM 48
#define ROWS  8                    // batch rows per workgroup
#define NTHR  (ROWS * T_DIM)       // 384 threads = 12 waves

#define LOG2E_F 1.4426950408889634f
#define LN2_F   0.6931471805599453f

// ---------------------------------------------------------------------------
// Prep: c_j = max_i trans[j,i];  Eexp[j,i] = exp(trans[j,i] - c_j)
// (runs once; use accurate expf since E is reused 511 times)
// ---------------------------------------------------------------------------
__global__ __launch_bounds__(64) void crf_prep(const float* __restrict__ trans,
                                               float* __restrict__ Eexp,
                                               float* __restrict__ cbuf) {
    int j = threadIdx.x;
    if (j >= T_DIM) return;
    float c = -INFINITY;
    #pragma unroll
    for (int i = 0; i < T_DIM; ++i) c = fmaxf(c, trans[j * T_DIM + i]);
    cbuf[j] = c;
    #pragma unroll
    for (int i = 0; i < T_DIM; ++i)
        Eexp[j * T_DIM + i] = expf(trans[j * T_DIM + i] - c);
}

// ---------------------------------------------------------------------------
// Main scan: per-WG 8 batch rows, thread = (b, j). Viterbi on VALU,
// log-sum-exp as exp-space f32 WMMA (16x16x4) on waves 0..2.
// ---------------------------------------------------------------------------
__global__ __launch_bounds__(NTHR) void crf_scan(
    const float* __restrict__ em, const float* __restrict__ mask,
    const float* __restrict__ trans, const float* __restrict__ start,
    const float* __restrict__ stop, const float* __restrict__ Eexp,
    const float* __restrict__ cbuf,
    float* __restrict__ alpha_out, float* __restrict__ best_out) {

    __shared__ float fv_lds[ROWS][T_DIM];
    __shared__ float vv_lds[ROWS][T_DIM];
    __shared__ float lg_lds[ROWS][T_DIM];   // ln(D) from WMMA waves
    __shared__ float p_lds[16][50];         // A-matrix tile, stride 50 kills conflicts
    __shared__ float red[ROWS][16];
    __shared__ float m_lds[ROWS];

    const int tid = threadIdx.x;
    const int b   = tid / T_DIM;
    const int j   = tid - b * T_DIM;
    const int gb  = blockIdx.x * ROWS + b;

    // zero whole A tile once (rows 8..15 stay zero -> upper half of D unused)
    {
        float* pf = &p_lds[0][0];
        for (int idx = tid; idx < 16 * 50; idx += NTHR) pf[idx] = 0.0f;
    }

    // per-thread constants
    const float c_j    = cbuf[j];
    const float stop_j = stop[j];
    float trow[T_DIM];                       // transitions[j][*] in registers
    #pragma unroll
    for (int i = 0; i < T_DIM; ++i) trow[i] = trans[j * T_DIM + i];

    // state init: s = 0
    float fv = start[j] + em[(size_t)gb * T_DIM + j];
    float vv = fv;

    // WMMA setup: waves 0..2 own j-chunk = wid. B-fragments of E are constant.
    const int lane  = tid & 31;
    const int wid   = __builtin_amdgcn_readfirstlane(tid >> 5);
    const int nn    = lane & 15;
    const int khalf = (lane >> 4) << 1;      // 0 for lanes 0-15, 2 for 16-31
    v2f bfr[12];
    if (wid < 3) {
        const int jj = wid * 16 + nn;        // N index = output tag
        #pragma unroll
        for (int kc = 0; kc < 12; ++kc) {    // B[k][n] = Eexp[jj][k]
            bfr[kc].x = Eexp[jj * T_DIM + 4 * kc + khalf + 0];
            bfr[kc].y = Eexp[jj * T_DIM + 4 * kc + khalf + 1];
        }
    }
    __syncthreads();

    for (int s = 1; s < S_LEN; ++s) {
        const size_t row = (size_t)(s * B_DIM + gb) * T_DIM;
        const float e    = em[row + j];                       // coalesced
        const float mval = mask[s * B_DIM + gb];
        if (s + 1 < S_LEN)
            __builtin_prefetch(&em[row + (size_t)B_DIM * T_DIM + j], 0, 1);

        fv_lds[b][j] = fv;
        vv_lds[b][j] = vv;
        __syncthreads();                                       // A

        // row max of fv: 48 -> 16 -> 1
        if (j < 16)
            red[b][j] = fmaxf(fmaxf(fv_lds[b][j], fv_lds[b][j + 16]),
                              fv_lds[b][j + 32]);              // v_max3
        __syncthreads();                                       // B
        if (j == 0) {
            float m = red[b][0];
            #pragma unroll
            for (int i = 1; i < 16; ++i) m = fmaxf(m, red[b][i]);
            m_lds[b] = m;
        }
        __syncthreads();                                       // C

        const float mb = m_lds[b];
        // fv - mb <= 0 and > -~1e3: raw v_exp_f32 is safe (output in (0,1])
        p_lds[b][j] = __builtin_amdgcn_exp2f((fv - mb) * LOG2E_F);

        // Viterbi: registers vs broadcast LDS reads
        float acc = -INFINITY;
        const float4* vrow = (const float4*)(&vv_lds[b][0]);
        #pragma unroll
        for (int i4 = 0; i4 < 12; ++i4) {
            float4 v = vrow[i4];
            acc = fmaxf(acc, v.x + trow[4 * i4 + 0]);
            acc = fmaxf(acc, v.y + trow[4 * i4 + 1]);
            acc = fmaxf(acc, v.z + trow[4 * i4 + 2]);
            acc = fmaxf(acc, v.w + trow[4 * i4 + 3]);
        }
        const float new_vv = e + acc;
        __syncthreads();                                       // D: p tile ready

        if (wid < 3) {                                         // uniform branch: EXEC all 1s
            v8f d = {0.f, 0.f, 0.f, 0.f, 0.f, 0.f, 0.f, 0.f};
            #pragma unroll
            for (int kc = 0; kc < 12; ++kc) {
                v2f a = *(const v2f*)(&p_lds[nn][4 * kc + khalf]); // A[m][k]
                d = __builtin_amdgcn_wmma_f32_16x16x4_f32(
                        false, a, false, bfr[kc], (short)0, d, false, false);
            }
            if (lane < 16) {                                   // M=r rows 0..7 valid
                // D in (exp(-~10), 48]: never denormal -> raw v_log_f32 (log2)
                #pragma unroll
                for (int r = 0; r < ROWS; ++r)
                    lg_lds[r][wid * 16 + lane] =
                        __builtin_amdgcn_logf(d[r]) * LN2_F;
            }
        }
        __syncthreads();                                       // E

        const float new_fv = e + mb + c_j + lg_lds[b][j];
        fv = fmaf(mval, new_fv - fv, fv);                      // blend, 2 VALU
        vv = fmaf(mval, new_vv - vv, vv);
    }

    // finalize: alpha = LSE_j(fv + stop), best = max_j(vv + stop)
    fv_lds[b][j] = fv + stop_j;
    vv_lds[b][j] = vv + stop_j;
    __syncthreads();
    if (j == 0) {
        float mx = -INFINITY, bm = -INFINITY;
        #pragma unroll
        for (int i = 0; i < T_DIM; ++i) {
            mx = fmaxf(mx, fv_lds[b][i]);
            bm = fmaxf(bm, vv_lds[b][i]);
        }
        float sm = 0.0f;
        #pragma unroll
        for (int i = 0; i < T_DIM; ++i)
            sm += __builtin_amdgcn_exp2f((fv_lds[b][i] - mx) * LOG2E_F);
        alpha_out[gb] = fmaf(__builtin_amdgcn_logf(sm), LN2_F, mx);
        best_out[gb]  = bm;
    }
}

// ---------------------------------------------------------------------------
// Gold path score (parallel over s within a block, one block per batch row)
// ---------------------------------------------------------------------------
__global__ __launch_bounds__(128) void crf_real(
    const float* __restrict__ em, const int* __restrict__ tags,
    const float* __restrict__ mask, const float* __restrict__ trans,
    const float* __restrict__ start, const float* __restrict__ stop,
    float* __restrict__ real_out) {
    __shared__ float rbuf[128];
    const int b = blockIdx.x;
    const int t = threadIdx.x;
    float sum = 0.0f;
    for (int s = 1 + t; s < S_LEN; s += 128) {
        int tp = tags[(s - 1) * B_DIM + b];
        int tc = tags[s * B_DIM + b];
        sum += mask[s * B_DIM + b] *
               (trans[tp * T_DIM + tc] + em[(size_t)(s * B_DIM + b) * T_DIM + tc]);
    }
    rbuf[t] = sum;
    __syncthreads();
    for (int off = 64; off > 0; off >>= 1) {
        if (t < off) rbuf[t] += rbuf[t + off];
        __syncthreads();
    }
    if (t == 0)
        real_out[b] = start[tags[b]] + rbuf[0] + stop[tags[(S_LEN - 1) * B_DIM + b]];
}

// ---------------------------------------------------------------------------
// loss = mean(alpha - real)
// ---------------------------------------------------------------------------
__global__ __launch_bounds__(256) void crf_loss(const float* __restrict__ real_v,
                                                const float* __restrict__ alpha_v,
                                                float* __restrict__ loss) {
    __shared__ float rbuf[256];
    float sum = 0.0f;
    for (int i = threadIdx.x; i < B_DIM; i += 256) sum += alpha_v[i] - real_v[i];
    rbuf[threadIdx.x] = sum;
    __syncthreads();
    for (int off = 128; off > 0; off >>= 1) {
        if (threadIdx.x < off) rbuf[threadIdx.x] += rbuf[threadIdx.x + off];
        __syncthreads();
    }
    if (threadIdx.x == 0) loss[0] = rbuf[0] / (float)B_DIM;
}

// ---------------------------------------------------------------------------
extern "C" void kernel_launch(void* const* d_in, const int* in_sizes, int n_in,
                              void* d_out, int out_size, void* d_ws, size_t ws_size,
                              hipStream_t stream) {
    const float* em    = (const float*)d_in[0];   // (S,B,T) f32
    const int*   tags  = (const int*)  d_in[1];   // (S,B)   i32
    const float* mask  = (const float*)d_in[2];   // (S,B)   f32
    const float* trans = (const float*)d_in[3];   // (T,T)   f32
    const float* start = (const float*)d_in[4];   // (T,)    f32
    const float* stop  = (const float*)d_in[5];   // (T,)    f32

    float* out       = (float*)d_out;             // [loss | real(B) | alpha(B) | best(B)]
    float* loss_out  = out;
    float* real_out  = out + 1;
    float* alpha_out = out + 1 + B_DIM;
    float* best_out  = out + 1 + 2 * B_DIM;

    float* Eexp = (float*)d_ws;                   // T*T floats
    float* cbuf = Eexp + T_DIM * T_DIM;           // T floats

    crf_prep<<<1, 64, 0, stream>>>(trans, Eexp, cbuf);
    crf_scan<<<B_DIM / ROWS, NTHR, 0, stream>>>(em, mask, trans, start, stop,
                                                Eexp, cbuf, alpha_out, best_out);
    crf_real<<<B_DIM, 128, 0, stream>>>(em, tags, mask, trans, start, stop, real_out);
    crf_loss<<<1, 256, 0, stream>>>(real_out, alpha_out, loss_out);
}